// gradient_loss_81088982548756
// MI455X (gfx1250) — compile-verified
//
#include <hip/hip_runtime.h>
#include <math.h>
#include <stdint.h>

#define IMG_H 512
#define IMG_W 512
#define TILE  32
#define NTHREADS 256

// LDS tile geometry (with row padding to spread banks)
#define IN_DIM 40   // TILE + 2*4 halo (blur2 + sobel1 + nms1)
#define IN_STR 41
#define HB_W   36   // horizontally-blurred width
#define HB_STR 37
#define BL_DIM 36   // blurred region (TILE + 2*2)
#define BL_STR 37
#define MG_DIM 34   // magnitude region (TILE + 2*1)
#define MG_STR 35

#define USE_ASYNC_LDS 1

typedef float v2f __attribute__((ext_vector_type(2)));
typedef float v8f __attribute__((ext_vector_type(8)));

// 1D gaussian (sigma=1, size=5), normalized: outer(w,w) == reference kernel
__device__ __forceinline__ float gw(int i) {
    const float W1[5] = {0.05448868454f, 0.24420134170f, 0.40261995376f,
                         0.24420134170f, 0.05448868454f};
    return W1[i];
}

__global__ __launch_bounds__(NTHREADS)
void canny_tile_kernel(const float* __restrict__ X, const float* __restrict__ Y,
                       float* __restrict__ partial)
{
    __shared__ float sIn[IN_DIM * IN_STR];   // raw input tile + halo 4
    __shared__ float sHb[IN_DIM * HB_STR];   // horizontally blurred
    __shared__ float sBl[BL_DIM * BL_STR];   // fully blurred (masked to image)
    __shared__ float sMg[MG_DIM * MG_STR];   // gradient magnitude (masked)
    __shared__ float sRed[8];

    const int tid    = threadIdx.x;
    const int tile_c = blockIdx.x * TILE;
    const int tile_r = blockIdx.y * TILE;
    const size_t base = (size_t)blockIdx.z * (size_t)(IMG_H * IMG_W);

    float thinA[4], oriA[4];
    float acc = 0.0f;

    for (int img = 0; img < 2; ++img) {
        const float* __restrict__ src = (img == 0) ? X : Y;
        const float* sbase = src + base;   // wave-uniform image base (SGPR pair)
        __syncthreads();  // fence LDS reuse between images

        // ---- Phase 1: input tile + halo(4) -> LDS, zero outside image ----
        // In-bounds lanes: async global->LDS DMA (no VGPR staging, ASYNCcnt).
        // Out-of-image halo lanes: plain ds_store of 0.
        for (int i = tid; i < IN_DIM * IN_DIM; i += NTHREADS) {
            const int r = i / IN_DIM, c = i - r * IN_DIM;
            const int gr = tile_r - 4 + r, gc = tile_c - 4 + c;
            if ((unsigned)gr < IMG_H && (unsigned)gc < IMG_W) {
                const unsigned voff = (unsigned)((gr * IMG_W + gc) * 4); // byte off, <16MB
#if USE_ASYNC_LDS
                const unsigned lds_addr =
                    (unsigned)(uintptr_t)(&sIn[r * IN_STR + c]); // flat low32 == LDS addr
                asm volatile("global_load_async_to_lds_b32 %0, %1, %2"
                             :
                             : "v"(lds_addr), "v"(voff), "s"(sbase)
                             : "memory");
#else
                sIn[r * IN_STR + c] = sbase[gr * IMG_W + gc];
#endif
                if (img == 0) __builtin_prefetch(&Y[base + (size_t)(gr * IMG_W + gc)], 0, 3);
            } else {
                sIn[r * IN_STR + c] = 0.0f;
            }
        }
#if USE_ASYNC_LDS
        asm volatile("s_wait_asynccnt 0x0" ::: "memory");
#endif
        __syncthreads();

        // ---- Phase 2a: horizontal 5-tap gaussian (40 rows x 36 cols) ----
        for (int i = tid; i < IN_DIM * HB_W; i += NTHREADS) {
            const int r = i / HB_W, c = i - r * HB_W;
            const float* p = &sIn[r * IN_STR + c];
            float s =        gw(0) * p[0];
            s = fmaf(gw(1), p[1], s);
            s = fmaf(gw(2), p[2], s);
            s = fmaf(gw(3), p[3], s);
            s = fmaf(gw(4), p[4], s);
            sHb[r * HB_STR + c] = s;
        }
        __syncthreads();

        // ---- Phase 2b: vertical 5-tap; zero for positions outside image ----
        for (int i = tid; i < BL_DIM * BL_DIM; i += NTHREADS) {
            const int r = i / BL_DIM, c = i - r * BL_DIM;
            const int gr = tile_r - 2 + r, gc = tile_c - 2 + c;
            float v = 0.0f;
            if ((unsigned)gr < IMG_H && (unsigned)gc < IMG_W) {
                const float* p = &sHb[r * HB_STR + c];
                float s =        gw(0) * p[0];
                s = fmaf(gw(1), p[1 * HB_STR], s);
                s = fmaf(gw(2), p[2 * HB_STR], s);
                s = fmaf(gw(3), p[3 * HB_STR], s);
                s = fmaf(gw(4), p[4 * HB_STR], s);
                v = s;
            }
            sBl[r * BL_STR + c] = v;
        }
        __syncthreads();

        // ---- Phase 3: sobel magnitude (masked to image; zero-pad semantics) ----
        for (int i = tid; i < MG_DIM * MG_DIM; i += NTHREADS) {
            const int r = i / MG_DIM, c = i - r * MG_DIM;
            const int gr = tile_r - 1 + r, gc = tile_c - 1 + c;
            float v = 0.0f;
            if ((unsigned)gr < IMG_H && (unsigned)gc < IMG_W) {
                const float* n = &sBl[r * BL_STR + c];
                const float n00 = n[0],          n01 = n[1],          n02 = n[2];
                const float n10 = n[BL_STR],                          n12 = n[BL_STR + 2];
                const float n20 = n[2 * BL_STR], n21 = n[2 * BL_STR + 1], n22 = n[2 * BL_STR + 2];
                const float gx = (n02 - n00) + 2.0f * (n12 - n10) + (n22 - n20);
                const float gy = (n20 - n00) + 2.0f * (n21 - n01) + (n22 - n02);
                v = sqrtf(fmaf(gx, gx, fmaf(gy, gy, 1e-12f)));
            }
            sMg[r * MG_STR + c] = v;
        }
        __syncthreads();

        // ---- Phase 4: orientation + NMS on the 32x32 core (4 px/thread) ----
        // Bit-packed (dr+1,dc+1) LUTs for the 8 gradient directions.
        const unsigned DR = 43265u;  // dr for bins 0..7: 0,-1,-1,-1,0,1,1,1
        const unsigned DC = 36890u;  // dc for bins 0..7: 1, 1, 0,-1,-1,-1,0,1
        for (int p4 = 0; p4 < 4; ++p4) {
            const int i  = tid + p4 * NTHREADS;
            const int rr = i >> 5, cc = i & 31;
            const float* n = &sBl[(rr + 1) * BL_STR + (cc + 1)];
            const float n00 = n[0],          n01 = n[1],          n02 = n[2];
            const float n10 = n[BL_STR],                          n12 = n[BL_STR + 2];
            const float n20 = n[2 * BL_STR], n21 = n[2 * BL_STR + 1], n22 = n[2 * BL_STR + 2];
            const float gx = (n02 - n00) + 2.0f * (n12 - n10) + (n22 - n20);
            const float gy = (n20 - n00) + 2.0f * (n21 - n01) + (n22 - n02);

            const float deg = atan2f(gy, gx) * 57.29577951308232f;
            const float bin = rintf((deg + 180.0f) * (1.0f / 45.0f)); // round-half-even, like jnp.round
            const float ori = bin * 45.0f;
            const int   ip  = ((int)bin) & 7;          // bin in [0,8] -> %8

            const int dr = (int)((DR >> (2 * ip)) & 3u) - 1;
            const int dc = (int)((DC >> (2 * ip)) & 3u) - 1;
            const int cidx  = (rr + 1) * MG_STR + (cc + 1);
            const int delta = dr * MG_STR + dc;

            const float magc = sMg[cidx];
            const float dpos = magc - sMg[cidx + delta]; // center - neighbor(dir)
            const float dneg = magc - sMg[cidx - delta]; // opposite dir == -offset
            const float thin = (fminf(dpos, dneg) > 0.0f) ? magc : 0.0f;

            if (img == 0) { thinA[p4] = thin; oriA[p4] = ori; }
            else          { acc += fabsf(thinA[p4] - thin) + fabsf(oriA[p4] - ori); }
        }
    }

    // ---- fp32 WMMA wave reduction: D = A(acc) x B(ones) sums across lanes ----
    // A(16x4): lane L<16 -> A[L][0]=acc, lane L>=16 -> A[L-16][2]=acc, rest 0.
    // D[i][j] = acc[i] + acc[i+16]; summing the 8 D VGPRs per lane then one
    // shfl_xor(16) yields the full 32-lane sum in fp32.
    v2f a; a.x = acc;  a.y = 0.0f;
    v2f b; b.x = 1.0f; b.y = 1.0f;
    v8f cz = {};
    v8f d = __builtin_amdgcn_wmma_f32_16x16x4_f32(
        /*neg_a=*/false, a, /*neg_b=*/false, b,
        /*c_mod=*/(short)0, cz, /*reuse_a=*/false, /*reuse_b=*/false);
    float s = ((d[0] + d[1]) + (d[2] + d[3])) + ((d[4] + d[5]) + (d[6] + d[7]));
    s += __shfl_xor(s, 16, 32);   // wave32 total (same in every lane)

    const int lane = tid & 31, wv = tid >> 5;
    if (lane == 0) sRed[wv] = s;
    __syncthreads();
    if (tid == 0) {
        float t = 0.0f;
        for (int w = 0; w < 8; ++w) t += sRed[w];  // fixed order: deterministic
        partial[((size_t)blockIdx.z * gridDim.y + blockIdx.y) * gridDim.x + blockIdx.x] = t;
    }
}

__global__ __launch_bounds__(NTHREADS)
void reduce_final_kernel(const float* __restrict__ partial, float* __restrict__ out, int n)
{
    __shared__ float sm[NTHREADS];
    const int t = threadIdx.x;
    float s = 0.0f;
    for (int i = t; i < n; i += NTHREADS) s += partial[i];  // fixed order
    sm[t] = s;
    __syncthreads();
    for (int off = NTHREADS / 2; off > 0; off >>= 1) {
        if (t < off) sm[t] += sm[t + off];
        __syncthreads();
    }
    if (t == 0) out[0] = sm[0] * (1.0f / (512.0f * 512.0f));
}

extern "C" void kernel_launch(void* const* d_in, const int* in_sizes, int n_in,
                              void* d_out, int out_size, void* d_ws, size_t ws_size,
                              hipStream_t stream)
{
    (void)n_in; (void)out_size; (void)ws_size;
    const float* X = (const float*)d_in[0];
    const float* Y = (const float*)d_in[1];
    float* partial = (float*)d_ws;

    const int B = in_sizes[0] / (IMG_H * IMG_W);   // 16
    dim3 grid(IMG_W / TILE, IMG_H / TILE, B);      // 16 x 16 x 16 = 4096 blocks

    canny_tile_kernel<<<grid, NTHREADS, 0, stream>>>(X, Y, partial);
    reduce_final_kernel<<<1, NTHREADS, 0, stream>>>(
        partial, (float*)d_out, B * (IMG_W / TILE) * (IMG_H / TILE));
}